// GCN_3006477107341
// MI455X (gfx1250) — compile-verified
//
#include <hip/hip_runtime.h>
#include <hip/hip_bf16.h>

typedef __attribute__((ext_vector_type(2))) float v2f;
typedef __attribute__((ext_vector_type(8))) float v8f;

#define FEATS 128

// ---------------------------------------------------------------------------
// Degree accumulation: deg_out[src[e]] += 1 ; deg_in[dst[e]] += 1
// ---------------------------------------------------------------------------
__global__ void gcn_degrees(const int* __restrict__ src, const int* __restrict__ dst,
                            float* __restrict__ dOut, float* __restrict__ dIn, int nE) {
    int e = blockIdx.x * blockDim.x + threadIdx.x;
    if (e >= nE) return;
    atomicAdd(&dOut[src[e]], 1.0f);
    atomicAdd(&dIn[dst[e]], 1.0f);
}

// deg -> clip(deg,1)^-0.5  (in place)
__global__ void gcn_norm(float* __restrict__ dOut, float* __restrict__ dIn, int n) {
    int i = blockIdx.x * blockDim.x + threadIdx.x;
    if (i >= n) return;
    dOut[i] = rsqrtf(fmaxf(dOut[i], 1.0f));
    dIn[i]  = rsqrtf(fmaxf(dIn[i], 1.0f));
}

// ---------------------------------------------------------------------------
// h = (x * invOut[:,None]) @ W1   via V_WMMA_F32_16X16X4_F32, fp32 in/out.
// One wave per 16x16 output tile; rowTile = wave>>3, colTile = wave&7.
// N % 16 == 0 and grid is exact, so EXEC is all ones (WMMA requirement).
// A-tile layout: lanes 0-15 -> K = {k,k+1}, lanes 16-31 -> K = {k+2,k+3}, M = lane&15.
// B-tile layout: mirror of A with N = lane&15.
// C/D layout: VGPR r -> M = r (lanes 0-15) / r+8 (lanes 16-31), N = lane&15.
// ---------------------------------------------------------------------------
__global__ void gcn_gemm1_wmma(const float* __restrict__ x, const float* __restrict__ W1,
                               const float* __restrict__ invOut, float* __restrict__ h) {
    const int lane    = threadIdx.x & 31;
    const int wave    = blockIdx.x * (blockDim.x >> 5) + (threadIdx.x >> 5);
    const int rowTile = wave >> 3;  // n/16 row tiles
    const int colTile = wave & 7;   // 128/16 col tiles
    const int l15     = lane & 15;
    const int khalf   = lane >> 4;  // 0 or 1

    const int   row  = rowTile * 16 + l15;   // A row (M) for this lane
    const int   nCol = colTile * 16 + l15;   // B/D column (N) for this lane
    const float s    = invOut[row];          // fold D_out^{-1/2} into A
    const float* __restrict__ arow = x + (size_t)row * FEATS;

    v8f c = {};
#pragma unroll
    for (int k = 0; k < FEATS; k += 4) {
        const int ka = k + 2 * khalf;
        v2f a, b;
        a.x = arow[ka]     * s;
        a.y = arow[ka + 1] * s;
        b.x = W1[(size_t)ka       * FEATS + nCol];
        b.y = W1[(size_t)(ka + 1) * FEATS + nCol];
        c = __builtin_amdgcn_wmma_f32_16x16x4_f32(
                /*neg_a=*/false, a, /*neg_b=*/false, b,
                /*c_mod=*/(short)0, c, /*reuse_a=*/false, /*reuse_b=*/false);
    }

#pragma unroll
    for (int r = 0; r < 8; ++r) {
        const int mm = (lane < 16) ? r : (r + 8);
        h[(size_t)(rowTile * 16 + mm) * FEATS + nCol] = c[r];
    }
}

// ---------------------------------------------------------------------------
// agg[dst[e],:] += h[src[e],:]  — 32 lanes per edge, float4 per lane,
// global_atomic_add_f32 (no return) into L2-resident agg.
// ---------------------------------------------------------------------------
__global__ void gcn_scatter_rows(const float* __restrict__ h, const int* __restrict__ src,
                                 const int* __restrict__ dst, float* __restrict__ agg, int nE) {
    const int tid = blockIdx.x * blockDim.x + threadIdx.x;
    const int e   = tid >> 5;
    if (e >= nE) return;
    const int f = (tid & 31) * 4;
    const float4 v = *(const float4*)(h + (size_t)src[e] * FEATS + f);
    float* p = agg + (size_t)dst[e] * FEATS + f;
    atomicAdd(p + 0, v.x);
    atomicAdd(p + 1, v.y);
    atomicAdd(p + 2, v.z);
    atomicAdd(p + 3, v.w);
}

// h = relu(agg * invIn[:,None] + b1)   (overwrites h)
__global__ void gcn_post1(const float* __restrict__ agg, const float* __restrict__ invIn,
                          const float* __restrict__ b1, float* __restrict__ h, int n) {
    const int i = blockIdx.x * blockDim.x + threadIdx.x;
    if (i >= n * FEATS) return;
    const int node = i >> 7;
    const int feat = i & (FEATS - 1);
    const float v = agg[i] * invIn[node] + b1[feat];
    h[i] = fmaxf(v, 0.0f);
}

// ---------------------------------------------------------------------------
// z[i] = invOut[i] * dot(h[i,:], W2)   — one wave per node, float4 per lane,
// shuffle reduction. n % 8 == 0 so the grid is exact.
// ---------------------------------------------------------------------------
__global__ void gcn_dot2(const float* __restrict__ h, const float* __restrict__ W2,
                         const float* __restrict__ invOut, float* __restrict__ z) {
    const int node = blockIdx.x * (blockDim.x >> 5) + (threadIdx.x >> 5);
    const int lane = threadIdx.x & 31;
    const float4 hv = *(const float4*)(h + (size_t)node * FEATS + lane * 4);
    const float4 wv = *(const float4*)(W2 + lane * 4);
    float sum = hv.x * wv.x + hv.y * wv.y + hv.z * wv.z + hv.w * wv.w;
#pragma unroll
    for (int off = 16; off > 0; off >>= 1) sum += __shfl_down(sum, off, 32);
    if (lane == 0) z[node] = sum * invOut[node];
}

// agg2[dst[e]] += z[src[e]]
__global__ void gcn_scatter_scalar(const float* __restrict__ z, const int* __restrict__ src,
                                   const int* __restrict__ dst, float* __restrict__ agg2, int nE) {
    const int e = blockIdx.x * blockDim.x + threadIdx.x;
    if (e >= nE) return;
    atomicAdd(&agg2[dst[e]], z[src[e]]);
}

// out[i] = agg2[i] * invIn[i] + b2[0]
__global__ void gcn_final(const float* __restrict__ agg2, const float* __restrict__ invIn,
                          const float* __restrict__ b2, float* __restrict__ out, int n) {
    const int i = blockIdx.x * blockDim.x + threadIdx.x;
    if (i >= n) return;
    out[i] = agg2[i] * invIn[i] + b2[0];
}

// ---------------------------------------------------------------------------
extern "C" void kernel_launch(void* const* d_in, const int* in_sizes, int n_in,
                              void* d_out, int out_size, void* d_ws, size_t ws_size,
                              hipStream_t stream) {
    const float* x   = (const float*)d_in[0];
    const int*   src = (const int*)d_in[1];
    const int*   dst = (const int*)d_in[2];
    const float* W1  = (const float*)d_in[3];
    const float* b1  = (const float*)d_in[4];
    const float* W2  = (const float*)d_in[5];
    const float* b2  = (const float*)d_in[6];
    float* out = (float*)d_out;

    const int n  = in_sizes[0] / FEATS;  // 100000 (divisible by 16 and 8)
    const int nE = in_sizes[1];          // 1600000

    // Workspace layout (floats): invOut[n] invIn[n] z[n] agg2[n] h[n*128] agg[n*128]
    float* f      = (float*)d_ws;
    float* invOut = f;
    float* invIn  = f + (size_t)n;
    float* z      = f + 2 * (size_t)n;
    float* agg2   = f + 3 * (size_t)n;
    float* h      = f + 4 * (size_t)n;
    float* agg    = h + (size_t)n * FEATS;

    // Re-zero accumulators every call (graph replay safe, capture-legal).
    hipMemsetAsync(invOut, 0, 2 * (size_t)n * sizeof(float), stream);
    hipMemsetAsync(agg2,   0, (size_t)n * sizeof(float), stream);
    hipMemsetAsync(agg,    0, (size_t)n * FEATS * sizeof(float), stream);

    gcn_degrees<<<(nE + 255) / 256, 256, 0, stream>>>(src, dst, invOut, invIn, nE);
    gcn_norm<<<(n + 255) / 256, 256, 0, stream>>>(invOut, invIn, n);

    // Layer 1 GEMM: (n/16) blocks * 8 waves = (n/16)*8 tiles, exact.
    gcn_gemm1_wmma<<<n / 16, 256, 0, stream>>>(x, W1, invOut, h);

    gcn_scatter_rows<<<(int)(((size_t)nE * 32 + 255) / 256), 256, 0, stream>>>(h, src, dst, agg, nE);
    gcn_post1<<<(int)(((size_t)n * FEATS + 255) / 256), 256, 0, stream>>>(agg, invIn, b1, h, n);

    // Layer 2: per-node dot with W2 (exact grid: n/8 blocks * 8 waves).
    gcn_dot2<<<n / 8, 256, 0, stream>>>(h, W2, invOut, z);
    gcn_scatter_scalar<<<(nE + 255) / 256, 256, 0, stream>>>(z, src, dst, agg2, nE);
    gcn_final<<<(n + 255) / 256, 256, 0, stream>>>(agg2, invIn, b2, out, n);
}